// DynamicGraphModel_23579370455152
// MI455X (gfx1250) — compile-verified
//
#include <hip/hip_runtime.h>
#include <hip/hip_bf16.h>

#define T_STEPS 20
#define D_INDIM 16
#define HDIM    128
#define GDIM    384   // 3*H
#define H_LD    132   // h tile LDS stride: 132 % 64 == 4 -> conflict-free b64 frag loads
#define A_LD    260   // concat tile LDS stride: 260 % 64 == 4
#define F_LD    136   // fused tile LDS stride

typedef __attribute__((ext_vector_type(2))) float v2f;
typedef __attribute__((ext_vector_type(8))) float v8f;

__device__ __forceinline__ v8f wmma_f32(v2f a, v2f b, v8f c) {
    // D = A(16x4) * B(4x16) + C, full fp32
    return __builtin_amdgcn_wmma_f32_16x16x4_f32(false, a, false, b, (short)0, c,
                                                 false, false);
}

__device__ __forceinline__ float sigmoidf(float x) {
    return 1.0f / (1.0f + __expf(-x));
}

// ---------------------------------------------------------------------------
// Kernel 0: pack weights as transposed adjacent-k pairs:
//   Wp[kh][g] = { W[g][2*kh], W[g][2*kh+1] }
// so a B fragment (rows k,k+1 at col g0+n) is ONE b64 load per lane,
// contiguous 128B per half-wave.
// ---------------------------------------------------------------------------
__global__ void pack_w_kernel(const float* __restrict__ W_hh,
                              const float* __restrict__ W_ih,
                              const float* __restrict__ W_fuse,
                              float* __restrict__ Wp_hh,   // [64][384] pairs
                              float* __restrict__ Wp_ih,   // [8][384] pairs
                              float* __restrict__ Wp_fuse) // [128][128] pairs
{
    int i = blockIdx.x * blockDim.x + threadIdx.x;
    if (i < 64 * GDIM) {                     // W_hh [384][128]
        int kh = i / GDIM, g = i - kh * GDIM;
        Wp_hh[2 * i]     = W_hh[g * HDIM + 2 * kh];
        Wp_hh[2 * i + 1] = W_hh[g * HDIM + 2 * kh + 1];
    }
    if (i < 8 * GDIM) {                      // W_ih [384][16]
        int kh = i / GDIM, g = i - kh * GDIM;
        Wp_ih[2 * i]     = W_ih[g * D_INDIM + 2 * kh];
        Wp_ih[2 * i + 1] = W_ih[g * D_INDIM + 2 * kh + 1];
    }
    if (i < 128 * HDIM) {                    // W_fuse [128][256]
        int kh = i >> 7, f = i & 127;
        Wp_fuse[2 * i]     = W_fuse[f * 256 + 2 * kh];
        Wp_fuse[2 * i + 1] = W_fuse[f * 256 + 2 * kh + 1];
    }
}

// ---------------------------------------------------------------------------
// Kernel 1: GRU via V_WMMA_F32_16X16X4_F32. One wave = one 16-node tile.
// h state double-buffered in LDS. B addressing is base-pointer + affine
// (kk * 2*GDIM) so every fragment load is base + constant immediate offset.
// ---------------------------------------------------------------------------
__global__ __launch_bounds__(128, 1) void gru_wmma_kernel(
    const float* __restrict__ x_seq,
    const v2f*  __restrict__ Wp_ih,    // [8][384]
    const v2f*  __restrict__ Wp_hh,    // [64][384]
    const float* __restrict__ b_ih,
    const float* __restrict__ b_hh,
    float* __restrict__ tout,          // [N][T][128]
    int N)
{
    __shared__ float sH[4][2][16 * H_LD];

    const int tid  = threadIdx.x;
    const int wave = tid >> 5;
    const int lane = tid & 31;
    const int tile0 = (blockIdx.x * 4 + wave) * 16;
    if (tile0 >= N) return;

    const int m  = lane & 15;   // A row / B,C,D column
    const int hf = lane >> 4;   // half-wave select

    // zero initial h buffer (per-wave private; in-wave LDS ordering suffices)
    for (int i = lane; i < 16 * H_LD; i += 32) sH[wave][0][i] = 0.0f;

    for (int t = 0; t < T_STEPS; ++t) {
        const float* hcur = sH[wave][t & 1];
        float*       hnxt = sH[wave][(t & 1) ^ 1];

        // x A-fragments (K = 16 -> 4 k-steps), kept in 8 VGPRs
        v2f xA[4];
        const float* xrow = x_seq + ((size_t)(tile0 + m) * T_STEPS + t) * D_INDIM;
#pragma unroll
        for (int kk = 0; kk < 4; ++kk)
            xA[kk] = *(const v2f*)&xrow[4 * kk + 2 * hf];

        // A-fragment base for h (LDS), constant 16B stride per k-step
        const float* ap = &hcur[m * H_LD + 2 * hf];

        for (int j = 0; j < 8; ++j) {          // 8 column tiles of H
            const int c0 = j * 16;
            const int gr = c0 + m;             // r-gate column in G

            // B base pointers: gate z at +HDIM, gate n at +2*HDIM elements
            const v2f* bih = Wp_ih + hf * GDIM + gr;
            const v2f* bhh = Wp_hh + hf * GDIM + gr;

            v8f R, Z, XN, HN;
            {
                float br = b_ih[gr]            + b_hh[gr];
                float bz = b_ih[HDIM + gr]     + b_hh[HDIM + gr];
                float bx = b_ih[2 * HDIM + gr];
                float bh = b_hh[2 * HDIM + gr];
#pragma unroll
                for (int i = 0; i < 8; ++i) { R[i] = br; Z[i] = bz; XN[i] = bx; HN[i] = bh; }
            }

            // x-projection contribution (K=16)
#pragma unroll
            for (int kk = 0; kk < 4; ++kk) {
                const v2f a  = xA[kk];
                const v2f br = bih[2 * GDIM * kk];
                const v2f bz = bih[2 * GDIM * kk + HDIM];
                const v2f bn = bih[2 * GDIM * kk + 2 * HDIM];
                R  = wmma_f32(a, br, R);
                Z  = wmma_f32(a, bz, Z);
                XN = wmma_f32(a, bn, XN);
            }
            // recurrent contribution (K=128), A-fragments streamed from LDS
#pragma unroll 8
            for (int kk = 0; kk < 32; ++kk) {
                const v2f a  = *(const v2f*)&ap[4 * kk];
                const v2f br = bhh[2 * GDIM * kk];
                const v2f bz = bhh[2 * GDIM * kk + HDIM];
                const v2f bn = bhh[2 * GDIM * kk + 2 * HDIM];
                R  = wmma_f32(a, br, R);
                Z  = wmma_f32(a, bz, Z);
                HN = wmma_f32(a, bn, HN);
            }

            // gates + state update + temporal_out store
#pragma unroll
            for (int i = 0; i < 8; ++i) {
                const int rowm = i + 8 * hf;   // D layout: M = i (+8 for hi half)
                float hold = hcur[rowm * H_LD + c0 + m];
                float r    = sigmoidf(R[i]);
                float z    = sigmoidf(Z[i]);
                float cand = tanhf(XN[i] + r * HN[i]);
                float hnew = (1.0f - z) * cand + z * hold;
                hnxt[rowm * H_LD + c0 + m] = hnew;
                float* op = &tout[((size_t)(tile0 + rowm) * T_STEPS + t) * HDIM + c0 + m];
                if (t == T_STEPS - 1) *op = hnew;              // reused -> cacheable
                else __builtin_nontemporal_store(hnew, op);    // 512MB stream -> NT
            }
        }
    }
}

// ---------------------------------------------------------------------------
// Kernel 2: zero aggregation buffers in workspace.
// ---------------------------------------------------------------------------
__global__ void zero_ws_kernel(float* __restrict__ agg, float* __restrict__ cnt, int N)
{
    size_t stride = (size_t)gridDim.x * blockDim.x;
    size_t i0 = (size_t)blockIdx.x * blockDim.x + threadIdx.x;
    size_t na = (size_t)N * HDIM;
    for (size_t i = i0; i < na; i += stride) agg[i] = 0.0f;
    for (size_t i = i0; i < (size_t)N; i += stride) cnt[i] = 0.0f;
}

// ---------------------------------------------------------------------------
// Kernel 3: edge scatter with heading mask. blockDim = 128 (feature dim).
// Mask uniform per edge -> no intra-wave divergence; f32 atomics land in L2
// (agg 25.6MB + current 25.6MB both L2-resident).
// ---------------------------------------------------------------------------
__global__ __launch_bounds__(128) void edge_kernel(
    const float* __restrict__ headings,
    const int*   __restrict__ ei,      // [2][E]
    const float* __restrict__ tout,    // current = tout[:, T-1, :]
    const int*   __restrict__ thr_ptr,
    float* __restrict__ agg,
    float* __restrict__ cnt,
    int E)
{
    const int f = threadIdx.x;
    const float thr = (float)(*thr_ptr);
    for (int e = blockIdx.x; e < E; e += gridDim.x) {
        const int s = ei[e];
        const int d = ei[E + e];
        float hd = fabsf(headings[s] - headings[d]);
        hd = fminf(hd, 360.0f - hd);
        if (hd <= thr) {
            float v = tout[((size_t)s * T_STEPS + (T_STEPS - 1)) * HDIM + f];
            atomicAdd(&agg[(size_t)d * HDIM + f], v);
            if (f == 0) atomicAdd(&cnt[d], 1.0f);
        }
    }
}

// ---------------------------------------------------------------------------
// Kernel 4: fused layer (WMMA, K=256) + ReLU + prediction head.
// Concat tile staged in LDS; affine base-pointer B addressing.
// ---------------------------------------------------------------------------
__global__ __launch_bounds__(128, 1) void fuse_wmma_kernel(
    const float* __restrict__ tout,
    const float* __restrict__ agg,
    const float* __restrict__ cnt,
    const v2f*  __restrict__ Wp_fuse,  // [128][128] pairs
    const float* __restrict__ b_fuse,
    const float* __restrict__ W_pred,  // [2][128]
    const float* __restrict__ b_pred,
    float* __restrict__ fused_out,     // [N][128]
    float* __restrict__ pred_out,      // [N][2]
    int N)
{
    __shared__ float sA[4][16 * A_LD];
    __shared__ float sF[4][16 * F_LD];

    const int tid  = threadIdx.x;
    const int wave = tid >> 5;
    const int lane = tid & 31;
    const int tile0 = (blockIdx.x * 4 + wave) * 16;
    if (tile0 >= N) return;

    const int m  = lane & 15;
    const int hf = lane >> 4;

    // Stage A = [current | neighbor_message] (16 x 256) into LDS
    float* sAw = sA[wave];
    for (int idx = lane; idx < 16 * 256; idx += 32) {
        const int r = idx >> 8, c = idx & 255;
        const int row = tile0 + r;
        float v;
        if (c < HDIM) {
            v = tout[((size_t)row * T_STEPS + (T_STEPS - 1)) * HDIM + c];
        } else {
            float inv = 1.0f / fmaxf(cnt[row], 1.0f);
            v = agg[(size_t)row * HDIM + (c - HDIM)] * inv;
        }
        sAw[r * A_LD + c] = v;
    }

    const float* ap = &sAw[m * A_LD + 2 * hf];
    float* sFw = sF[wave];
    for (int j = 0; j < 8; ++j) {       // 8 output tiles of FD=128
        const int f0 = j * 16;
        const v2f* bp = Wp_fuse + hf * HDIM + f0 + m;
        v8f C;
        {
            float b = b_fuse[f0 + m];
#pragma unroll
            for (int i = 0; i < 8; ++i) C[i] = b;
        }
#pragma unroll 8
        for (int kk = 0; kk < 64; ++kk) {
            const v2f a = *(const v2f*)&ap[4 * kk];
            const v2f b = bp[2 * HDIM * kk];
            C = wmma_f32(a, b, C);
        }
#pragma unroll
        for (int i = 0; i < 8; ++i) {
            const int rowm = i + 8 * hf;
            float v = fmaxf(C[i], 0.0f);
            fused_out[(size_t)(tile0 + rowm) * HDIM + f0 + m] = v;
            sFw[rowm * F_LD + f0 + m] = v;
        }
    }

    // prediction head: lane -> (node m, output hf); 2 outputs per node
    float acc = b_pred[hf];
#pragma unroll 8
    for (int f = 0; f < HDIM; ++f)
        acc += sFw[m * F_LD + f] * W_pred[hf * HDIM + f];
    pred_out[(size_t)(tile0 + m) * 2 + hf] = acc;
}

// ---------------------------------------------------------------------------
extern "C" void kernel_launch(void* const* d_in, const int* in_sizes, int n_in,
                              void* d_out, int out_size, void* d_ws, size_t ws_size,
                              hipStream_t stream)
{
    const float* x_seq    = (const float*)d_in[0];
    const float* headings = (const float*)d_in[1];
    const float* W_ih     = (const float*)d_in[2];
    const float* W_hh     = (const float*)d_in[3];
    const float* b_ih     = (const float*)d_in[4];
    const float* b_hh     = (const float*)d_in[5];
    const float* W_fuse   = (const float*)d_in[6];
    const float* b_fuse   = (const float*)d_in[7];
    const float* W_pred   = (const float*)d_in[8];
    const float* b_pred   = (const float*)d_in[9];
    const int*   edges    = (const int*)d_in[10];
    const int*   thr      = (const int*)d_in[11];

    const int N = in_sizes[1];
    const int E = in_sizes[10] / 2;

    // d_out layout: predictions (N*2) | temporal_out (N*T*H) | fused (N*H)
    float* out   = (float*)d_out;
    float* pred  = out;
    float* tout  = out + (size_t)N * 2;
    float* fused = tout + (size_t)N * T_STEPS * HDIM;

    // workspace: agg (N*128) | cnt (N) | Wp_hh | Wp_ih | Wp_fuse
    float* agg     = (float*)d_ws;
    float* cnt     = agg + (size_t)N * HDIM;
    float* Wp_hh   = cnt + N;                      // 64*384*2 floats
    float* Wp_ih   = Wp_hh + 64 * GDIM * 2;        // 8*384*2 floats
    float* Wp_fuse = Wp_ih + 8 * GDIM * 2;         // 128*128*2 floats

    const int tiles  = (N + 15) / 16;
    const int blocks = (tiles + 3) / 4;

    hipLaunchKernelGGL(pack_w_kernel, dim3((64 * GDIM + 255) / 256), dim3(256),
                       0, stream, W_hh, W_ih, W_fuse, Wp_hh, Wp_ih, Wp_fuse);
    hipLaunchKernelGGL(gru_wmma_kernel, dim3(blocks), dim3(128), 0, stream,
                       x_seq, (const v2f*)Wp_ih, (const v2f*)Wp_hh, b_ih, b_hh, tout, N);
    hipLaunchKernelGGL(zero_ws_kernel, dim3(1024), dim3(256), 0, stream, agg, cnt, N);
    hipLaunchKernelGGL(edge_kernel, dim3(8192), dim3(128), 0, stream,
                       headings, edges, tout, thr, agg, cnt, E);
    hipLaunchKernelGGL(fuse_wmma_kernel, dim3(blocks), dim3(128), 0, stream,
                       tout, agg, cnt, (const v2f*)Wp_fuse, b_fuse, W_pred, b_pred,
                       fused, pred, N);
}